// causal_message_passing_rdm_91173565760140
// MI455X (gfx1250) — compile-verified
//
#include <hip/hip_runtime.h>
#include <hip/hip_bf16.h>
#include <climits>

// ---------------------------------------------------------------------------
// Problem constants (match reference)
// ---------------------------------------------------------------------------
#define N_NODES 20000
#define N_EDGES 160000
#define IN_DIM  256
#define OUT_DIM 256
#define NHEAD   8
#define DK      32
#define RSQRT_DK 0.17677669529663687f  // 1/sqrt(32)

typedef float v2f __attribute__((ext_vector_type(2)));
typedef float v8f __attribute__((ext_vector_type(8)));

// ---------------------------------------------------------------------------
// Order-preserving float<->int map for atomicMax-based segment max
// ---------------------------------------------------------------------------
__device__ __forceinline__ int f2ord(float f) {
    int i = __float_as_int(f);
    return (i >= 0) ? i : (i ^ 0x7fffffff);
}
__device__ __forceinline__ float ord2f(int o) {
    int i = (o >= 0) ? o : (o ^ 0x7fffffff);
    return __int_as_float(i);
}

// ---------------------------------------------------------------------------
// Kernel 1: fp32 WMMA GEMM  Out[M,256] = X[M,256] @ W[256,256] + bias
// One 16x16 output tile per wave, K-loop in steps of 4 via v_wmma_f32_16x16x4_f32
// grid: (M/16, 2), block: 256 (8 waves -> 16 column tiles of 16)
// ---------------------------------------------------------------------------
__global__ __launch_bounds__(256) void gemm_wmma_f32(
    const float* __restrict__ X, const float* __restrict__ W,
    const float* __restrict__ bias, float* __restrict__ Out)
{
    const int lane = threadIdx.x & 31;
    const int wave = threadIdx.x >> 5;
    const int rowBase = blockIdx.x * 16;
    const int colBase = (blockIdx.y * 8 + wave) * 16;
    const int half = lane >> 4;   // K-half selector for A/B fragments
    const int r    = lane & 15;   // row (A) / col (B,C) within tile

    v8f c = {};
    const float* __restrict__ xrow = X + (size_t)(rowBase + r) * IN_DIM;

#pragma unroll 8
    for (int kk = 0; kk < IN_DIM; kk += 4) {
        const int ka = kk + 2 * half;
        // A fragment: 16x4 f32. lanes 0-15 hold K=kk,kk+1; lanes 16-31 hold K=kk+2,kk+3
        v2f a;
        a.x = xrow[ka + 0];
        a.y = xrow[ka + 1];
        // B fragment: 4x16 f32, mirrored half-wave K split, row-striped across lanes
        v2f b;
        b.x = W[(size_t)(ka + 0) * OUT_DIM + colBase + r];
        b.y = W[(size_t)(ka + 1) * OUT_DIM + colBase + r];
        c = __builtin_amdgcn_wmma_f32_16x16x4_f32(
                /*neg_a=*/false, a, /*neg_b=*/false, b,
                /*c_mod=*/(short)0, c, /*reuse_a=*/false, /*reuse_b=*/false);
    }

    const float bb = bias[colBase + r];
#pragma unroll
    for (int i = 0; i < 8; ++i) {
        const int row = rowBase + i + 8 * half;   // C/D layout: VGPR i -> M=i / M=i+8
        Out[(size_t)row * OUT_DIM + colBase + r] = c[i] + bb;
    }
}

// ---------------------------------------------------------------------------
// Kernel 2: masked_k[n,h,f] = sum_d k[n,h,d] * cau_filter[ct[n],h,d,f]
// one thread per (n,h,f); filter matrices (96 KB total) live in cache
// ---------------------------------------------------------------------------
__global__ __launch_bounds__(256) void maskedk_kernel(
    const float* __restrict__ k, const float* __restrict__ cfil,
    const int* __restrict__ ct, float* __restrict__ mk)
{
    const int gid = blockIdx.x * blockDim.x + threadIdx.x;
    if (gid >= N_NODES * OUT_DIM) return;
    const int n = gid >> 8;
    const int h = (gid >> 5) & 7;
    const int f = gid & 31;
    const float* __restrict__ krow = k + (size_t)n * OUT_DIM + h * DK;
    const float* __restrict__ M = cfil + ((size_t)(ct[n] * NHEAD + h)) * (DK * DK) + f;
    float acc = 0.f;
#pragma unroll
    for (int d = 0; d < DK; ++d) acc = fmaf(krow[d], M[d * DK], acc);
    mk[gid] = acc;
}

// ---------------------------------------------------------------------------
// Kernel 3 (per etype): val = v @ msg[e],  cval = (v + time_emb) @ msg_cau[e]
// ---------------------------------------------------------------------------
__global__ __launch_bounds__(256) void valcval_kernel(
    const float* __restrict__ v, const float* __restrict__ msg_e,
    const float* __restrict__ msgc_e, const float* __restrict__ temb,
    float* __restrict__ val, float* __restrict__ cval)
{
    const int gid = blockIdx.x * blockDim.x + threadIdx.x;
    if (gid >= N_NODES * OUT_DIM) return;
    const int n = gid >> 8;
    const int h = (gid >> 5) & 7;
    const int f = gid & 31;
    const float* __restrict__ vrow = v + (size_t)n * OUT_DIM + h * DK;
    const float* __restrict__ te   = temb + h * DK;
    const float* __restrict__ M    = msg_e  + (size_t)h * (DK * DK) + f;
    const float* __restrict__ Mc   = msgc_e + (size_t)h * (DK * DK) + f;
    float a = 0.f, b = 0.f;
#pragma unroll
    for (int d = 0; d < DK; ++d) {
        const float vv = vrow[d];
        a = fmaf(vv, M[d * DK], a);
        b = fmaf(vv + te[d], Mc[d * DK], b);
    }
    val[gid]  = a;
    cval[gid] = b;
}

// ---------------------------------------------------------------------------
// Zero / init helpers
// ---------------------------------------------------------------------------
__global__ void zero_f32(float* __restrict__ p, int n) {
    const int gid = blockIdx.x * blockDim.x + threadIdx.x;
    if (gid < n) p[gid] = 0.f;
}
__global__ void init_softmax_state(int* __restrict__ mord, int* __restrict__ cmord,
                                   float* __restrict__ s, float* __restrict__ cs, int n) {
    const int gid = blockIdx.x * blockDim.x + threadIdx.x;
    if (gid < n) { mord[gid] = INT_MIN; cmord[gid] = INT_MIN; s[gid] = 0.f; cs[gid] = 0.f; }
}

// ---------------------------------------------------------------------------
// Kernel 4 (per etype): attention logits + segment max (ordered-int atomicMax)
// one thread per (edge, head)
// ---------------------------------------------------------------------------
__global__ __launch_bounds__(256) void logits_kernel(
    const float* __restrict__ q, const float* __restrict__ k,
    const float* __restrict__ mk,
    const int* __restrict__ src, const int* __restrict__ dst,
    const float* __restrict__ pri_e, const float* __restrict__ pric_e,
    float* __restrict__ att, float* __restrict__ catt,
    int* __restrict__ mord, int* __restrict__ cmord)
{
    const int gid = blockIdx.x * blockDim.x + threadIdx.x;
    if (gid >= N_EDGES * NHEAD) return;
    const int edge = gid >> 3;
    const int h = gid & 7;
    const int sn = src[edge];
    const int dn = dst[edge];

    const float4* __restrict__ qv = (const float4*)(q  + (size_t)dn * OUT_DIM + h * DK);
    const float4* __restrict__ kv = (const float4*)(k  + (size_t)sn * OUT_DIM + h * DK);
    const float4* __restrict__ mv = (const float4*)(mk + (size_t)sn * OUT_DIM + h * DK);

    float da = 0.f, dc = 0.f;
#pragma unroll
    for (int i = 0; i < DK / 4; ++i) {
        const float4 qq = qv[i], kk = kv[i], mm = mv[i];
        da = fmaf(qq.x, kk.x, fmaf(qq.y, kk.y, fmaf(qq.z, kk.z, fmaf(qq.w, kk.w, da))));
        dc = fmaf(qq.x, mm.x, fmaf(qq.y, mm.y, fmaf(qq.z, mm.z, fmaf(qq.w, mm.w, dc))));
    }
    const float a = da * pri_e[h]  * RSQRT_DK;
    const float c = dc * pric_e[h] * RSQRT_DK;
    att[gid]  = a;
    catt[gid] = c;
    atomicMax(&mord[dn * NHEAD + h],  f2ord(a));
    atomicMax(&cmord[dn * NHEAD + h], f2ord(c));
}

// ---------------------------------------------------------------------------
// Kernel 5 (per etype): exp(logit - max) and segment sum
// ---------------------------------------------------------------------------
__global__ __launch_bounds__(256) void expsum_kernel(
    const int* __restrict__ dst,
    float* __restrict__ att, float* __restrict__ catt,
    const int* __restrict__ mord, const int* __restrict__ cmord,
    float* __restrict__ s, float* __restrict__ cs)
{
    const int gid = blockIdx.x * blockDim.x + threadIdx.x;
    if (gid >= N_EDGES * NHEAD) return;
    const int edge = gid >> 3;
    const int h = gid & 7;
    const int dn = dst[edge];
    const int seg = dn * NHEAD + h;

    const float w = __expf(att[gid] - ord2f(mord[seg]));
    att[gid] = w;
    atomicAdd(&s[seg], w);

    const float cw = __expf(catt[gid] - ord2f(cmord[seg]));
    catt[gid] = cw;
    atomicAdd(&cs[seg], cw);
}

// ---------------------------------------------------------------------------
// Kernel 6 (per etype): weighted scatter-add of (a*val + ca*comb_pri*cval)
// into the shared accumulator (mean over etypes folded in at the end)
// ---------------------------------------------------------------------------
__global__ __launch_bounds__(256) void scatter_kernel(
    const int* __restrict__ src, const int* __restrict__ dst,
    const float* __restrict__ att, const float* __restrict__ catt,
    const float* __restrict__ s, const float* __restrict__ cs,
    const float* __restrict__ val, const float* __restrict__ cval,
    const float* __restrict__ comb_e, float* __restrict__ acc)
{
    const int gid = blockIdx.x * blockDim.x + threadIdx.x;
    if (gid >= N_EDGES * NHEAD) return;
    const int edge = gid >> 3;
    const int h = gid & 7;
    const int sn = src[edge];
    const int dn = dst[edge];
    const int seg = dn * NHEAD + h;

    const float a  = att[gid]  / s[seg];
    const float ca = catt[gid] / cs[seg];

    const float4* __restrict__ vv = (const float4*)(val  + (size_t)sn * OUT_DIM + h * DK);
    const float4* __restrict__ cv = (const float4*)(cval + (size_t)sn * OUT_DIM + h * DK);
    const float4* __restrict__ cp = (const float4*)(comb_e + h * DK);
    float* __restrict__ dacc = acc + (size_t)dn * OUT_DIM + h * DK;

#pragma unroll
    for (int i = 0; i < DK / 4; ++i) {
        const float4 v4 = vv[i], c4 = cv[i], p4 = cp[i];
        atomicAdd(&dacc[4 * i + 0], fmaf(ca * p4.x, c4.x, a * v4.x));
        atomicAdd(&dacc[4 * i + 1], fmaf(ca * p4.y, c4.y, a * v4.y));
        atomicAdd(&dacc[4 * i + 2], fmaf(ca * p4.z, c4.z, a * v4.z));
        atomicAdd(&dacc[4 * i + 3], fmaf(ca * p4.w, c4.w, a * v4.w));
    }
}

// ---------------------------------------------------------------------------
// Kernel 7: out = relu(acc / 3)
// ---------------------------------------------------------------------------
__global__ void final_kernel(const float* __restrict__ acc, float* __restrict__ out, int n)
{
    const int gid = blockIdx.x * blockDim.x + threadIdx.x;
    if (gid < n) out[gid] = fmaxf(acc[gid] * (1.f / 3.f), 0.f);
}

// ---------------------------------------------------------------------------
// Host launcher
// ---------------------------------------------------------------------------
extern "C" void kernel_launch(void* const* d_in, const int* in_sizes, int n_in,
                              void* d_out, int out_size, void* d_ws, size_t ws_size,
                              hipStream_t stream)
{
    const float* x    = (const float*)d_in[0];
    const float* Wk   = (const float*)d_in[1];
    const float* bk   = (const float*)d_in[2];
    const float* Wq   = (const float*)d_in[3];
    const float* bq   = (const float*)d_in[4];
    const float* Wv   = (const float*)d_in[5];
    const float* bv   = (const float*)d_in[6];
    const float* pri  = (const float*)d_in[7];   // [3,8]
    const float* msg  = (const float*)d_in[8];   // [3,8,32,32]
    const float* pric = (const float*)d_in[9];   // [3,8]
    const float* msgc = (const float*)d_in[10];  // [3,8,32,32]
    const float* comb = (const float*)d_in[11];  // [3,8,32]
    const float* cfil = (const float*)d_in[12];  // [3,8,32,32]
    const float* temb = (const float*)d_in[13];  // [8,1,32]
    const int*   ct   = (const int*)d_in[14];    // [N]
    const int*   src  = (const int*)d_in[15];    // [3,E]
    const int*   dst  = (const int*)d_in[16];    // [3,E]
    float* out = (float*)d_out;

    // workspace layout (fp32 unless noted)
    const size_t NF = (size_t)N_NODES * OUT_DIM;   // 5.12M
    const size_t EH = (size_t)N_EDGES * NHEAD;     // 1.28M
    const size_t NH = (size_t)N_NODES * NHEAD;     // 160K
    float* ws   = (float*)d_ws;
    float* kbuf = ws;            // NF
    float* qbuf = kbuf + NF;     // NF
    float* vbuf = qbuf + NF;     // NF
    float* mk   = vbuf + NF;     // NF
    float* val  = mk   + NF;     // NF (reused per etype)
    float* cval = val  + NF;     // NF (reused per etype)
    float* acc  = cval + NF;     // NF (accumulates all etypes)
    float* att  = acc  + NF;     // EH (reused per etype)
    float* catt = att  + EH;     // EH
    float* sbuf = catt + EH;     // NH
    float* csbuf= sbuf + NH;     // NH
    int*   mord = (int*)(csbuf + NH); // NH
    int*   cmord= mord + NH;          // NH

    const int B = 256;
    const dim3 gGemm(N_NODES / 16, 2);

    // dense projections via fp32 WMMA
    gemm_wmma_f32<<<gGemm, B, 0, stream>>>(x, Wk, bk, kbuf);
    gemm_wmma_f32<<<gGemm, B, 0, stream>>>(x, Wq, bq, qbuf);
    gemm_wmma_f32<<<gGemm, B, 0, stream>>>(x, Wv, bv, vbuf);

    // node-level causal key transform
    maskedk_kernel<<<(int)((NF + B - 1) / B), B, 0, stream>>>(kbuf, cfil, ct, mk);

    // clear accumulator
    zero_f32<<<(int)((NF + B - 1) / B), B, 0, stream>>>(acc, (int)NF);

    for (int e = 0; e < 3; ++e) {
        const int*   src_e  = src + (size_t)e * N_EDGES;
        const int*   dst_e  = dst + (size_t)e * N_EDGES;
        const float* msg_e  = msg  + (size_t)e * NHEAD * DK * DK;
        const float* msgc_e = msgc + (size_t)e * NHEAD * DK * DK;
        const float* pri_e  = pri  + (size_t)e * NHEAD;
        const float* pric_e = pric + (size_t)e * NHEAD;
        const float* comb_e = comb + (size_t)e * NHEAD * DK;

        valcval_kernel<<<(int)((NF + B - 1) / B), B, 0, stream>>>(
            vbuf, msg_e, msgc_e, temb, val, cval);

        init_softmax_state<<<(int)((NH + B - 1) / B), B, 0, stream>>>(
            mord, cmord, sbuf, csbuf, (int)NH);

        logits_kernel<<<(int)((EH + B - 1) / B), B, 0, stream>>>(
            qbuf, kbuf, mk, src_e, dst_e, pri_e, pric_e, att, catt, mord, cmord);

        expsum_kernel<<<(int)((EH + B - 1) / B), B, 0, stream>>>(
            dst_e, att, catt, mord, cmord, sbuf, csbuf);

        scatter_kernel<<<(int)((EH + B - 1) / B), B, 0, stream>>>(
            src_e, dst_e, att, catt, sbuf, csbuf, val, cval, comb_e, acc);
    }

    final_kernel<<<(int)((NF + B - 1) / B), B, 0, stream>>>(acc, out, (int)NF);
}